// OptimizedWRGCNLayer_85890755985720
// MI455X (gfx1250) — compile-verified
//
#include <hip/hip_runtime.h>
#include <stdint.h>

typedef __attribute__((ext_vector_type(2))) float v2f;
typedef __attribute__((ext_vector_type(8))) float v8f;

#define D      256   // D_IN == D_OUT
#define LDA    260   // padded LDS row stride (floats): banks 4m+k, conflict-free
#define BLK_M  32    // rows (edges or entities) per workgroup
#define KC     32    // K-chunk staged in LDS
#define NCHUNK (D / KC)

// ---- CDNA5 async global->LDS (ASYNCcnt path) --------------------------------
__device__ __forceinline__ uint32_t lds_off(const void* p) {
  // shared-aperture flat address: ADDR[31:0] == workgroup-relative LDS offset
  return (uint32_t)(uintptr_t)p;
}
__device__ __forceinline__ void async_ld_b128(uint32_t dst_lds, const void* src) {
  asm volatile("global_load_async_to_lds_b128 %0, %1, off"
               :: "v"(dst_lds), "v"(src) : "memory");
}
__device__ __forceinline__ void wait_async0() {
  asm volatile("s_wait_asynccnt 0x0" ::: "memory");
}

// ---- stage one K-chunk of W into packed LDS ---------------------------------
// Packed layout: buf[(kk>>2)*1024 + n*4 + (kk&3)], kk = k within chunk.
// B-frag (4x16) for a lane then reads 2 contiguous floats -> one ds_load_b64.
__device__ __forceinline__ void load_B_chunk(const float* __restrict__ W, int kc,
                                             float* buf, int tid) {
  const int n0 = (tid & 63) * 4;   // 64 threads span n = 0..255 in float4s
  const int g0 = tid >> 6;         // k4-group 0..3 (+4 on second pass)
#pragma unroll
  for (int gg = 0; gg < 2; ++gg) {
    const int g = g0 + gg * 4;                 // 0..7
    const int kb = kc * KC + g * 4;
    float4 r0 = *(const float4*)(W + (size_t)(kb + 0) * D + n0);
    float4 r1 = *(const float4*)(W + (size_t)(kb + 1) * D + n0);
    float4 r2 = *(const float4*)(W + (size_t)(kb + 2) * D + n0);
    float4 r3 = *(const float4*)(W + (size_t)(kb + 3) * D + n0);
    float4* d = (float4*)(buf + g * (D * 4) + n0 * 4);   // 4x4 register transpose
    d[0] = make_float4(r0.x, r1.x, r2.x, r3.x);
    d[1] = make_float4(r0.y, r1.y, r2.y, r3.y);
    d[2] = make_float4(r0.z, r1.z, r2.z, r3.z);
    d[3] = make_float4(r0.w, r1.w, r2.w, r3.w);
  }
}

// ---- one wave: 16-row M-tile x 4 N-tiles over a KC chunk --------------------
// A-frag (16x4 f32): lanes 0-15 hold K={k,k+1}, lanes 16-31 hold K={k+2,k+3}.
__device__ __forceinline__ void gemm_chunk_wave(const float* Bbuf, const float* arow,
                                                int n_group, int lane, v8f acc[4]) {
  const int hf = lane >> 4, l15 = lane & 15, koff = hf * 2;
#pragma unroll
  for (int kk = 0; kk < KC; kk += 4) {
    v2f a;
    a.x = arow[kk + koff];
    a.y = arow[kk + koff + 1];
    const float* bg = Bbuf + (kk >> 2) * (D * 4);
#pragma unroll
    for (int i = 0; i < 4; ++i) {
      const int nb = (n_group * 4 + i) * 16;
      const float* bp = bg + (nb + l15) * 4 + koff;
      v2f b;
      b.x = bp[0];
      b.y = bp[1];
      acc[i] = __builtin_amdgcn_wmma_f32_16x16x4_f32(
          false, a, false, b, (short)0, acc[i], false, false);
    }
  }
}

// ---- out = X @ Wself (initializes every element of out) --------------------
__global__ __launch_bounds__(256) void self_gemm_kernel(
    const float* __restrict__ X, const float* __restrict__ Wself,
    float* __restrict__ out, int N)
{
  __shared__ float As[BLK_M * LDA];
  __shared__ float Bs[2][KC * D];
  const int tid = threadIdx.x;
  const int mb  = blockIdx.x * BLK_M;
  {
    const int rr = tid >> 3, c0 = (tid & 7) * 32;
    int row = mb + rr; if (row >= N) row = 0;
    const float* g = X + (size_t)row * D + c0;
    const uint32_t l = lds_off(As + rr * LDA + c0);
#pragma unroll
    for (int i = 0; i < 8; ++i) async_ld_b128(l + i * 16, g + i * 4);
  }
  load_B_chunk(Wself, 0, Bs[0], tid);
  wait_async0();
  __syncthreads();

  const int wave = tid >> 5, lane = tid & 31;
  const int m_tile = wave >> 2, n_group = wave & 3;
  const float* arow = As + (m_tile * 16 + (lane & 15)) * LDA;
  v8f acc[4] = {};
  for (int kc = 0; kc < NCHUNK; ++kc) {
    if (kc + 1 < NCHUNK) load_B_chunk(Wself, kc + 1, Bs[(kc + 1) & 1], tid);
    gemm_chunk_wave(Bs[kc & 1], arow + kc * KC, n_group, lane, acc);
    __syncthreads();
  }

  const int hf = lane >> 4, l15 = lane & 15;
#pragma unroll
  for (int i = 0; i < 4; ++i) {
    const int nb = (n_group * 4 + i) * 16;
#pragma unroll
    for (int v = 0; v < 8; ++v) {
      const int row = mb + m_tile * 16 + v + hf * 8;   // C/D layout: VGPR v -> M=v / v+8
      if (row < N) out[(size_t)row * D + nb + l15] = acc[i][v];
    }
  }
}

// ---- per-relation: (X[src] @ W_r) * ew -> atomic scatter into out[tgt] ------
__global__ __launch_bounds__(256) void edge_msg_kernel(
    const float* __restrict__ X, const float* __restrict__ relW,
    const int* __restrict__ edge_index, const float* __restrict__ edge_w,
    float* __restrict__ out, int N, int E)
{
  __shared__ float As[BLK_M * LDA];
  __shared__ float Bs[2][KC * D];
  __shared__ int   srcS[BLK_M];
  __shared__ int   tgtS[BLK_M];
  __shared__ float ewS[BLK_M];
  const int tid = threadIdx.x;
  const int r   = blockIdx.y;
  const long long eb = (long long)blockIdx.x * BLK_M;
  const int* srcIdx = edge_index + (long long)r * 2 * E;
  const int* tgtIdx = srcIdx + E;
  if (tid < BLK_M) {
    const long long e = eb + tid;
    const bool valid = e < E;
    int s = valid ? srcIdx[e] : 0;
    if (s < 0 || s >= N) s = 0;
    int t = valid ? tgtIdx[e] : -1;
    if (t < 0 || t >= N) t = -1;
    srcS[tid] = s;
    tgtS[tid] = t;
    ewS[tid]  = valid ? edge_w[(long long)r * E + e] : 0.0f;
  }
  __syncthreads();
  {
    const int rr = tid >> 3, c0 = (tid & 7) * 32;
    const float* g = X + (size_t)srcS[rr] * D + c0;   // gathered row
    const uint32_t l = lds_off(As + rr * LDA + c0);
#pragma unroll
    for (int i = 0; i < 8; ++i) async_ld_b128(l + i * 16, g + i * 4);
  }
  const float* Wr = relW + (size_t)r * D * D;
  load_B_chunk(Wr, 0, Bs[0], tid);
  wait_async0();
  __syncthreads();

  const int wave = tid >> 5, lane = tid & 31;
  const int m_tile = wave >> 2, n_group = wave & 3;
  const float* arow = As + (m_tile * 16 + (lane & 15)) * LDA;
  v8f acc[4] = {};
  for (int kc = 0; kc < NCHUNK; ++kc) {
    if (kc + 1 < NCHUNK) load_B_chunk(Wr, kc + 1, Bs[(kc + 1) & 1], tid);
    gemm_chunk_wave(Bs[kc & 1], arow + kc * KC, n_group, lane, acc);
    __syncthreads();
  }

  const int hf = lane >> 4, l15 = lane & 15;
#pragma unroll
  for (int i = 0; i < 4; ++i) {
    const int nb = (n_group * 4 + i) * 16;
#pragma unroll
    for (int v = 0; v < 8; ++v) {
      const int rowl = m_tile * 16 + v + hf * 8;
      const int t = tgtS[rowl];
      if (t >= 0)
        atomicAdd(&out[(size_t)t * D + nb + l15], acc[i][v] * ewS[rowl]);
    }
  }
}

// ---- presence-gated bias ----------------------------------------------------
__global__ void zero_ws_kernel(unsigned* ws, int nwords) {
  const int i = blockIdx.x * blockDim.x + threadIdx.x;
  if (i < nwords) ws[i] = 0u;
}

__global__ void mark_presence_kernel(const int* __restrict__ edge_index,
                                     unsigned* __restrict__ ws,
                                     int N, int E, int R, int wpr)
{
  const long long idx = (long long)blockIdx.x * blockDim.x + threadIdx.x;
  if (idx >= (long long)R * E) return;
  const int r = (int)(idx / E);
  const long long e = idx - (long long)r * E;
  const int t = edge_index[(long long)r * 2 * E + E + e];
  if (t >= 0 && t < N)
    atomicOr(&ws[(long long)r * wpr + (t >> 5)], 1u << (t & 31));
}

__global__ void bias_add_kernel(const unsigned* __restrict__ ws,
                                const float* __restrict__ bias,
                                float* __restrict__ out, int N, int R, int wpr)
{
  const long long idx = (long long)blockIdx.x * blockDim.x + threadIdx.x;
  if (idx >= (long long)N * 64) return;
  const int n = (int)(idx >> 6);
  const int c = (int)(idx & 63) * 4;
  const int word = n >> 5, bit = n & 31;
  float4 a = make_float4(0.f, 0.f, 0.f, 0.f);
  bool any = false;
  for (int r = 0; r < R; ++r) {
    if ((ws[(long long)r * wpr + word] >> bit) & 1u) {
      const float* b = bias + (long long)r * D + c;
      a.x += b[0]; a.y += b[1]; a.z += b[2]; a.w += b[3];
      any = true;
    }
  }
  if (any) {
    float* o = out + (long long)n * D + c;
    o[0] += a.x; o[1] += a.y; o[2] += a.z; o[3] += a.w;
  }
}

extern "C" void kernel_launch(void* const* d_in, const int* in_sizes, int n_in,
                              void* d_out, int out_size, void* d_ws, size_t ws_size,
                              hipStream_t stream) {
  const float* X    = (const float*)d_in[0];   // [N, 256]
  const float* Wr   = (const float*)d_in[1];   // [R, 256, 256]
  const float* Ws   = (const float*)d_in[2];   // [256, 256]
  const float* bias = (const float*)d_in[3];   // [R, 256]
  const int*   ei   = (const int*)d_in[4];     // [R, 2, E]
  const float* ew   = (const float*)d_in[5];   // [R, E]
  float* out = (float*)d_out;

  const int N = in_sizes[0] / D;
  const int R = in_sizes[3] / D;
  const int E = (R > 0) ? in_sizes[5] / R : 0;

  // 1) out = X @ self_weight (writes every element of out)
  const int mblocks = (N + BLK_M - 1) / BLK_M;
  self_gemm_kernel<<<mblocks, 256, 0, stream>>>(X, Ws, out, N);

  // 2) edge messages with fused scale + atomic scatter-add
  dim3 egrid((unsigned)((E + BLK_M - 1) / BLK_M), (unsigned)R);
  edge_msg_kernel<<<egrid, 256, 0, stream>>>(X, Wr, ei, ew, out, N, E);

  // 3) presence-gated bias via bitmap in workspace
  const int wpr = (N + 31) / 32;
  const size_t needed = (size_t)R * wpr * sizeof(unsigned);
  if (ws_size >= needed) {
    unsigned* ws = (unsigned*)d_ws;
    const int nwords = R * wpr;
    zero_ws_kernel<<<(nwords + 255) / 256, 256, 0, stream>>>(ws, nwords);
    const long long marks = (long long)R * E;
    mark_presence_kernel<<<(unsigned)((marks + 255) / 256), 256, 0, stream>>>(ei, ws, N, E, R, wpr);
    const long long bthreads = (long long)N * 64;
    bias_add_kernel<<<(unsigned)((bthreads + 255) / 256), 256, 0, stream>>>(ws, bias, out, N, R, wpr);
  }
}